// GraphAttentionSparse_88502096101457
// MI455X (gfx1250) — compile-verified
//
#include <hip/hip_runtime.h>
#include <hip/hip_bf16.h>

typedef float v2f __attribute__((ext_vector_type(2)));
typedef float v8f __attribute__((ext_vector_type(8)));

#define IN_F 128
#define OUT_F 64
#define LEAKY_ALPHA 0.2f

// ---------------------------------------------------------------------------
// GEMM: Wh = x @ W via v_wmma_f32_16x16x4_f32 (full fp32 precision).
// Block = 256 threads (8 waves), each wave computes a 16-row x 64-col tile.
// W staged in LDS in k-pair-interleaved layout so each B fragment is a
// single aligned ds_load_b64 (no VGPR repacking movs in the inner loop):
//   sW2[(k/2)*2*OUT_F + 2*n + 0] = W[k  ][n]
//   sW2[(k/2)*2*OUT_F + 2*n + 1] = W[k+1][n]
// ---------------------------------------------------------------------------
__global__ __launch_bounds__(256) void gat_gemm_wmma(
    const float* __restrict__ x, const float* __restrict__ W,
    float* __restrict__ Wh, int n) {
  __shared__ float sW2[IN_F * OUT_F];

  // Cooperative interleaved fill: 4096 k-pairs*cols, 16 per thread.
  for (int i = threadIdx.x; i < (IN_F / 2) * OUT_F; i += blockDim.x) {
    const int kp = i >> 6;        // k-pair index (k = 2*kp)
    const int c  = i & 63;        // column
    v2f p = { W[(2 * kp) * OUT_F + c], W[(2 * kp + 1) * OUT_F + c] };
    *(v2f*)(sW2 + kp * (2 * OUT_F) + 2 * c) = p;
  }
  __syncthreads();

  const int lane  = threadIdx.x & 31;
  const int wave  = threadIdx.x >> 5;
  const int nidx  = lane & 15;   // M index (A) / N index (B,C)
  const int khalf = lane >> 4;   // K-half selector

  const int rowbase = (blockIdx.x * 8 + wave) * 16;
  if (rowbase >= n) return;  // wave-uniform branch: EXEC stays all-1 inside

  int row = rowbase + nidx;
  if (row >= n) row = n - 1;  // clamp loads; stores predicated below
  const float* xrow = x + (long)row * IN_F;

  v8f acc0 = {}, acc1 = {}, acc2 = {}, acc3 = {};

  for (int k0 = 0; k0 < IN_F; k0 += 4) {
    const int kb = k0 + 2 * khalf;   // even
    // A fragment: lane holds x[row][kb], x[row][kb+1] (contiguous float2)
    v2f a = *(const v2f*)(xrow + kb);
    // B fragments: single b64 LDS read each (pair-interleaved layout)
    const float* wb = sW2 + (kb >> 1) * (2 * OUT_F) + 2 * nidx;
    v2f b0 = *(const v2f*)(wb + 0);
    v2f b1 = *(const v2f*)(wb + 32);
    v2f b2 = *(const v2f*)(wb + 64);
    v2f b3 = *(const v2f*)(wb + 96);
    acc0 = __builtin_amdgcn_wmma_f32_16x16x4_f32(false, a, false, b0, (short)0, acc0, false, false);
    acc1 = __builtin_amdgcn_wmma_f32_16x16x4_f32(false, a, false, b1, (short)0, acc1, false, false);
    acc2 = __builtin_amdgcn_wmma_f32_16x16x4_f32(false, a, false, b2, (short)0, acc2, false, false);
    acc3 = __builtin_amdgcn_wmma_f32_16x16x4_f32(false, a, false, b3, (short)0, acc3, false, false);
  }

  // C/D layout: VGPR r, lane l -> row = rowbase + r + 8*(l>>4), col = (l&15)
#pragma unroll
  for (int r = 0; r < 8; ++r) {
    const int gr = rowbase + r + 8 * khalf;
    if (gr < n) {
      float* o = Wh + (long)gr * OUT_F + nidx;
      o[0]  = acc0[r];
      o[16] = acc1[r];
      o[32] = acc2[r];
      o[48] = acc3[r];
    }
  }
}

// ---------------------------------------------------------------------------
// Per-node: e_left/e_right = Wh . a_left/a_right; init e_max / e_sum.
// ---------------------------------------------------------------------------
__global__ void gat_node_ev(const float* __restrict__ Wh,
                            const float* __restrict__ a_left,
                            const float* __restrict__ a_right,
                            float* __restrict__ e_left, float* __restrict__ e_right,
                            float* __restrict__ e_max, float* __restrict__ e_sum, int n) {
  const int i = blockIdx.x * blockDim.x + threadIdx.x;
  if (i >= n) return;
  const float4* w = (const float4*)(Wh + (long)i * OUT_F);
  float sl = 0.f, sr = 0.f;
#pragma unroll
  for (int j = 0; j < OUT_F / 4; ++j) {
    float4 v  = w[j];
    float4 al = ((const float4*)a_left)[j];
    float4 ar = ((const float4*)a_right)[j];
    sl += v.x * al.x + v.y * al.y + v.z * al.z + v.w * al.w;
    sr += v.x * ar.x + v.y * ar.y + v.z * ar.z + v.w * ar.w;
  }
  e_left[i]  = sl;
  e_right[i] = sr;
  e_max[i]   = -3.402823466e38f;
  e_sum[i]   = 0.f;
}

__global__ void gat_zero_out(float* __restrict__ out, int n4) {
  const int i = blockIdx.x * blockDim.x + threadIdx.x;
  if (i < n4) ((float4*)out)[i] = make_float4(0.f, 0.f, 0.f, 0.f);
}

__device__ __forceinline__ float leaky(float e) {
  return e > 0.f ? e : LEAKY_ALPHA * e;
}

// Monotonic float max via signed/unsigned int atomics (IEEE, no NaNs).
__device__ __forceinline__ void atomicMaxFloat(float* addr, float v) {
  if (v >= 0.f)
    atomicMax((int*)addr, __float_as_int(v));
  else
    atomicMin((unsigned int*)addr, __float_as_uint(v));
}

// ---------------------------------------------------------------------------
// Edge pass 1: segment max of leaky(e_left[src] + e_right[dst]) by src.
// ---------------------------------------------------------------------------
__global__ void gat_edge_max(const int* __restrict__ ei,
                             const float* __restrict__ e_left,
                             const float* __restrict__ e_right,
                             float* __restrict__ e_max, int E) {
  const int e = blockIdx.x * blockDim.x + threadIdx.x;
  if (e >= E) return;
  const int s = ei[e];
  const int d = ei[E + e];
  atomicMaxFloat(&e_max[s], leaky(e_left[s] + e_right[d]));
}

// ---------------------------------------------------------------------------
// Edge pass 2: e_exp = exp(e - max[src]); segment sum by src.
// ---------------------------------------------------------------------------
__global__ void gat_edge_expsum(const int* __restrict__ ei,
                                const float* __restrict__ e_left,
                                const float* __restrict__ e_right,
                                const float* __restrict__ e_max,
                                float* __restrict__ e_exp,
                                float* __restrict__ e_sum, int E) {
  const int e = blockIdx.x * blockDim.x + threadIdx.x;
  if (e >= E) return;
  const int s = ei[e];
  const int d = ei[E + e];
  const float v  = leaky(e_left[s] + e_right[d]);
  const float ex = __expf(v - e_max[s]);
  e_exp[e] = ex;
  atomicAdd(&e_sum[s], ex);
}

// ---------------------------------------------------------------------------
// Edge pass 3: out[src] += (e_exp/(sum+eps)) * Wh[dst].  64 lanes per edge,
// so src/dst/e_exp/e_sum loads are wave-uniform (broadcast) and Wh/out
// accesses are fully coalesced; everything is L2-resident (192MB).
// ---------------------------------------------------------------------------
__global__ void gat_edge_scatter(const int* __restrict__ ei,
                                 const float* __restrict__ e_exp,
                                 const float* __restrict__ e_sum,
                                 const float* __restrict__ Wh,
                                 float* __restrict__ out, int E) {
  const long gid = (long)blockIdx.x * blockDim.x + threadIdx.x;
  const long e = gid >> 6;
  const int  f = (int)(gid & 63);
  if (e >= E) return;
  const int s = ei[e];
  const int d = ei[E + e];
  const float alpha = e_exp[e] / (e_sum[s] + 1e-9f);
  atomicAdd(&out[(long)s * OUT_F + f], alpha * Wh[(long)d * OUT_F + f]);
}

__global__ void gat_relu(float* __restrict__ out, int n4) {
  const int i = blockIdx.x * blockDim.x + threadIdx.x;
  if (i >= n4) return;
  float4 v = ((float4*)out)[i];
  v.x = fmaxf(v.x, 0.f);
  v.y = fmaxf(v.y, 0.f);
  v.z = fmaxf(v.z, 0.f);
  v.w = fmaxf(v.w, 0.f);
  ((float4*)out)[i] = v;
}

extern "C" void kernel_launch(void* const* d_in, const int* in_sizes, int n_in,
                              void* d_out, int out_size, void* d_ws, size_t ws_size,
                              hipStream_t stream) {
  const float* x       = (const float*)d_in[0];
  const float* W       = (const float*)d_in[1];
  const float* a_left  = (const float*)d_in[2];
  const float* a_right = (const float*)d_in[3];
  const int*   ei      = (const int*)d_in[4];   // [2, E] int32
  float* out = (float*)d_out;

  const int n = in_sizes[0] / IN_F;
  const int E = in_sizes[4] / 2;

  // Workspace layout
  float* Wh      = (float*)d_ws;                         // n * OUT_F
  float* e_left  = Wh + (size_t)n * OUT_F;               // n
  float* e_right = e_left + n;                           // n
  float* e_max   = e_right + n;                          // n
  float* e_sum   = e_max + n;                            // n
  float* e_exp   = e_sum + n;                            // E

  // 1) Wh = x @ W (WMMA fp32)
  gat_gemm_wmma<<<(n + 127) / 128, 256, 0, stream>>>(x, W, Wh, n);

  // 2) per-node scores + init
  gat_node_ev<<<(n + 255) / 256, 256, 0, stream>>>(Wh, a_left, a_right,
                                                   e_left, e_right, e_max, e_sum, n);

  // 3) zero output accumulator (d_out is poisoned by harness)
  const int n4 = n * OUT_F / 4;
  gat_zero_out<<<(n4 + 255) / 256, 256, 0, stream>>>(out, n4);

  // 4) segment softmax (3 atomic passes) + weighted scatter
  gat_edge_max<<<(E + 255) / 256, 256, 0, stream>>>(ei, e_left, e_right, e_max, E);
  gat_edge_expsum<<<(E + 255) / 256, 256, 0, stream>>>(ei, e_left, e_right, e_max,
                                                       e_exp, e_sum, E);
  const long scatter_threads = (long)E * OUT_F;
  gat_edge_scatter<<<(unsigned)((scatter_threads + 255) / 256), 256, 0, stream>>>(
      ei, e_exp, e_sum, Wh, out, E);

  // 5) final relu
  gat_relu<<<(n4 + 255) / 256, 256, 0, stream>>>(out, n4);
}